// VonMisesDistribution_9818295238983
// MI455X (gfx1250) — compile-verified
//
#include <hip/hip_runtime.h>
#include <math.h>
#include <stdint.h>

// ---------------------------------------------------------------------------
// Problem constants (fixed by setup_inputs)
// ---------------------------------------------------------------------------
constexpr int   Bsz   = 16384;   // batch
constexpr int   ENC   = 1024;    // encoder dim (K)
constexpr int   NLAT  = 256;     // latent dim (N)
constexpr float KAPPA = 100.0f;
constexpr float BN_EPS = 1e-4f;
constexpr int   MU_ELEMS = Bsz * NLAT;

// Workspace layout (float offsets)
constexpr size_t WS_MU   = 0;                       // [Bsz*NLAT] mu = data@W^T + b
constexpr size_t WS_SUM  = (size_t)MU_ELEMS;        // [NLAT] column sums
constexpr size_t WS_SQ   = WS_SUM + NLAT;           // [NLAT] column sums of squares
constexpr size_t WS_MEAN = WS_SQ + NLAT;            // [NLAT]
constexpr size_t WS_RSTD = WS_MEAN + NLAT;          // [NLAT]
constexpr size_t WS_KLD  = WS_RSTD + NLAT;          // [1] scalar vMF KLD

typedef float v2f __attribute__((ext_vector_type(2)));
typedef float v8f __attribute__((ext_vector_type(8)));

// ---------------------------------------------------------------------------
// CDNA5 async global->LDS copy (ASYNCcnt-tracked, no VGPR data movement)
// ---------------------------------------------------------------------------
__device__ inline void async_load_b128(uint32_t lds_off, const float* gptr) {
  asm volatile("global_load_async_to_lds_b128 %0, %1, off"
               :: "v"(lds_off), "v"((uint64_t)(uintptr_t)gptr)
               : "memory");
}
__device__ inline void wait_asynccnt0() {
  asm volatile("s_wait_asynccnt 0x0" ::: "memory");
}

// ---------------------------------------------------------------------------
// Kernel 0: zero the BN accumulators (ws is poisoned / stale between calls)
// ---------------------------------------------------------------------------
__global__ void init_stats_kernel(float* __restrict__ ws) {
  const int t = threadIdx.x;            // 512 threads
  if (t < 2 * NLAT) ws[WS_SUM + t] = 0.0f;
}

// ---------------------------------------------------------------------------
// Kernel 1: fp32 WMMA GEMM  mu = A @ W^T + b  (+ per-column partial sums)
//   grid.x = Bsz/32 blocks, 512 threads = 16 wave32s
//   wave w computes TWO 16x16 tiles: (m0, n0=16w) and (m0+16, n0=16w),
//   reusing each B fragment across both -> 2 WMMAs per B float2 load.
// ---------------------------------------------------------------------------
__global__ __launch_bounds__(512) void gemm_bn_partial_kernel(
    const float* __restrict__ A,      // [Bsz, ENC]
    const float* __restrict__ Wm,     // [NLAT, ENC]
    const float* __restrict__ bias,   // [NLAT]
    float* __restrict__ ws)
{
  __shared__ __align__(16) float lds_a[32 * 128];   // 16 KB A-strip chunk

  float* __restrict__ mu = ws + WS_MU;

  const int m0   = blockIdx.x * 32;
  const int tid  = threadIdx.x;
  const int wave = tid >> 5;
  const int lane = tid & 31;
  const int n0   = wave * 16;
  const int col  = lane & 15;           // A row (within tile) AND B/C column
  const int hi   = lane >> 4;           // 0: lanes 0-15, 1: lanes 16-31
  const int koff = hi * 2;              // K sub-offset per ISA 16x4 fp32 layout

  v8f acc0 = {0.f, 0.f, 0.f, 0.f, 0.f, 0.f, 0.f, 0.f};
  v8f acc1 = {0.f, 0.f, 0.f, 0.f, 0.f, 0.f, 0.f, 0.f};

  // cooperative A-chunk mapping: 512 threads x 2 float4 = 32x128 floats
  const int q0   = tid * 2;
  const int row0 = q0 >> 5;             // 0..31
  const int c0   = (q0 & 31) * 4;       // 0..124
  const int q1   = q0 + 1;
  const int row1 = q1 >> 5;
  const int c1   = (q1 & 31) * 4;

  const uint32_t lds_base = (uint32_t)(uintptr_t)&lds_a[0];
  const float* __restrict__ wbase = &Wm[(size_t)(n0 + col) * ENC];

  for (int kc = 0; kc < ENC; kc += 128) {
    __syncthreads();                                   // readers of prev chunk done
    async_load_b128(lds_base + (uint32_t)(row0 * 128 + c0) * 4u,
                    &A[(size_t)(m0 + row0) * ENC + kc + c0]);
    async_load_b128(lds_base + (uint32_t)(row1 * 128 + c1) * 4u,
                    &A[(size_t)(m0 + row1) * ENC + kc + c1]);
    if (kc + 128 < ENC)                                // pull next chunk toward WGP
      __builtin_prefetch(&A[(size_t)(m0 + row0) * ENC + kc + 128 + c0], 0, 3);
    wait_asynccnt0();                                  // own portions landed in LDS
    __syncthreads();                                   // all portions visible

    const float* __restrict__ wrow = wbase + kc;
    #pragma unroll 8
    for (int kk = 0; kk < 128; kk += 4) {
      // B fragment (4x16 fp32): lane n holds column n; lanes 0-15 K={kk,kk+1},
      // lanes 16-31 K={kk+2,kk+3}; shared by both M tiles
      const float2 bv = *reinterpret_cast<const float2*>(&wrow[kk + koff]);
      v2f b; b.x = bv.x; b.y = bv.y;
      // A fragments (16x4 fp32): lanes 0-15 M=lane K={kk,kk+1}; lanes16-31 K+2
      const float* ap0 = &lds_a[col * 128 + kk + koff];
      const float* ap1 = &lds_a[(16 + col) * 128 + kk + koff];
      v2f a0; a0.x = ap0[0]; a0.y = ap0[1];
      v2f a1; a1.x = ap1[0]; a1.y = ap1[1];
      acc0 = __builtin_amdgcn_wmma_f32_16x16x4_f32(
          false, a0, false, b, (short)0, acc0, false, false);
      acc1 = __builtin_amdgcn_wmma_f32_16x16x4_f32(
          false, a1, false, b, (short)0, acc1, false, false);
    }
  }

  // C/D layout: VGPR r -> row = hi*8 + r, column = col
  const float bb = bias[n0 + col];
  float v0[8], v1[8];
  float sum = 0.f, sq = 0.f;
  #pragma unroll
  for (int r = 0; r < 8; ++r) {
    const float a = acc0[r] + bb;
    const float c = acc1[r] + bb;
    v0[r] = a; v1[r] = c;
    sum += a + c;
    sq  += a * a + c * c;
  }
  // lane L and lane L+16 hold complementary row-halves of the same column
  sum += __shfl_xor(sum, 16, 32);
  sq  += __shfl_xor(sq, 16, 32);
  if (hi == 0) {
    atomicAdd(&ws[WS_SUM + n0 + col], sum);
    atomicAdd(&ws[WS_SQ  + n0 + col], sq);
  }
  #pragma unroll
  for (int r = 0; r < 8; ++r) {
    const int row = hi * 8 + r;
    mu[(size_t)(m0 + row)      * NLAT + n0 + col] = v0[r];
    mu[(size_t)(m0 + 16 + row) * NLAT + n0 + col] = v1[r];
  }
}

// ---------------------------------------------------------------------------
// Kernel 2: finalize BN stats; thread 0 also evaluates the vMF KLD constant
// ---------------------------------------------------------------------------
__device__ double log_iv_series(double v, double x) {
  const double lx2 = log(x * 0.5);
  double lse = v * lx2 - lgamma(v + 1.0);            // m = 0 term
  for (int m = 1; m < 300; ++m) {
    const double t = (v + 2.0 * m) * lx2
                     - lgamma((double)m + 1.0) - lgamma((double)m + v + 1.0);
    if (t > lse) lse = t + log1p(exp(lse - t));
    else         lse = lse + log1p(exp(t - lse));
  }
  return lse;
}

__global__ void finalize_stats_kernel(float* __restrict__ ws) {
  const int n = threadIdx.x;            // 256 threads
  const float invB = 1.0f / (float)Bsz;
  const float mean = ws[WS_SUM + n] * invB;
  const float var  = ws[WS_SQ + n] * invB - mean * mean;
  ws[WS_MEAN + n] = mean;
  ws[WS_RSTD + n] = rsqrtf(var + BN_EPS);
  if (n == 0) {
    const double d  = (double)NLAT;
    const double l0 = log_iv_series(d * 0.5,       (double)KAPPA);
    const double l1 = log_iv_series(d * 0.5 + 1.0, (double)KAPPA);
    const double ratio = exp(l1 - l0);
    const double kld = (double)KAPPA * ratio + d * log((double)KAPPA) * 0.5
                       - l0 - lgamma(d * 0.5 + 1.0) - d * log(2.0) * 0.5;
    ws[WS_KLD] = (float)kld;
  }
}

// ---------------------------------------------------------------------------
// Kernel 3: BN -> softplus -> von Mises (Best-Fisher rejection) -> wrap
// ---------------------------------------------------------------------------
__device__ inline uint32_t mix32(uint32_t x) {
  x ^= x >> 17; x *= 0xed5ad4bbu;
  x ^= x >> 11; x *= 0xac4c1b51u;
  x ^= x >> 15; x *= 0x31848babu;
  x ^= x >> 14; return x;
}
__device__ inline float u01(uint32_t h) {            // uniform in (0,1)
  return ((float)(h >> 8) + 0.5f) * (1.0f / 16777216.0f);
}

__global__ __launch_bounds__(256) void bn_softplus_vm_sample_kernel(
    const float* __restrict__ ws,
    const float* __restrict__ gamma,
    const float* __restrict__ beta,
    float* __restrict__ out)
{
  const int gid = blockIdx.x * 256 + threadIdx.x;    // < MU_ELEMS
  const int n   = gid & (NLAT - 1);

  const float muv  = ws[WS_MU + gid];
  const float mean = ws[WS_MEAN + n];
  const float rstd = ws[WS_RSTD + n];
  const float xbn  = gamma[n] * (muv - mean) * rstd + beta[n];
  const float loc  = (xbn > 20.0f) ? xbn : log1pf(__expf(xbn));

  // Best-Fisher proposal constants (constant-folded for kappa = 100)
  const float kappa = KAPPA;
  const float tau = 1.0f + sqrtf(1.0f + 4.0f * kappa * kappa);
  const float rho = (tau - sqrtf(2.0f * tau)) / (2.0f * kappa);
  const float r   = (1.0f + rho * rho) / (2.0f * rho);

  float xs = 0.0f;
  bool done = false;
  for (int it = 0; it < 64 && !done; ++it) {
    const uint32_t h0 = mix32(0x12345u ^ (uint32_t)gid ^ ((uint32_t)it * 0x9E3779B9u));
    const uint32_t h1 = mix32(h0 + 0x85ebca6bu);
    const uint32_t h2 = mix32(h1 + 0xc2b2ae35u);
    const float u1 = u01(h0), u2 = u01(h1), u3 = u01(h2);
    const float z = __cosf(3.14159265358979f * u1);
    const float f = (1.0f + r * z) / (r + z);
    const float c = kappa * (r - f);
    const bool accept = (c * (2.0f - c) - u2 > 0.0f) ||
                        (__logf(c / u2) + 1.0f - c >= 0.0f);
    if (accept) {
      const float fc = fminf(fmaxf(f, -1.0f), 1.0f);
      xs = (u3 < 0.5f ? -1.0f : 1.0f) * acosf(fc);
      done = true;
    }
  }
  const float PI_F = 3.14159265358979f;
  out[gid] = fmodf(xs + PI_F + loc, 2.0f * PI_F) - PI_F;

  if (gid < Bsz) out[(size_t)MU_ELEMS + gid] = ws[WS_KLD];   // constant KLD tail
}

// ---------------------------------------------------------------------------
// Launcher
// ---------------------------------------------------------------------------
extern "C" void kernel_launch(void* const* d_in, const int* in_sizes, int n_in,
                              void* d_out, int out_size, void* d_ws, size_t ws_size,
                              hipStream_t stream) {
  (void)in_sizes; (void)n_in; (void)out_size; (void)ws_size;
  const float* data  = (const float*)d_in[0];   // [Bsz, ENC]
  const float* Wm    = (const float*)d_in[1];   // [NLAT, ENC]
  const float* bias  = (const float*)d_in[2];   // [NLAT]
  const float* gamma = (const float*)d_in[3];   // [NLAT]
  const float* beta  = (const float*)d_in[4];   // [NLAT]
  float* ws  = (float*)d_ws;
  float* out = (float*)d_out;

  init_stats_kernel<<<1, 512, 0, stream>>>(ws);
  gemm_bn_partial_kernel<<<Bsz / 32, 512, 0, stream>>>(data, Wm, bias, ws);
  finalize_stats_kernel<<<1, NLAT, 0, stream>>>(ws);
  bn_softplus_vm_sample_kernel<<<MU_ELEMS / 256, 256, 0, stream>>>(ws, gamma, beta, out);
}